// MiniMemory_37288906064320
// MI455X (gfx1250) — compile-verified
//
#include <hip/hip_runtime.h>
#include <hip/hip_bf16.h>

#define BB 8
#define TT 4096
#define DD 2048

typedef __attribute__((ext_vector_type(2))) float v2f;
typedef __attribute__((ext_vector_type(8))) float v8f;

// ---------------------------------------------------------------------------
// Kernel 0: pack w_mass/w_decay into a [D][16] B-matrix image (cols 0,1 live,
// rest zero) so WMMA B-fragment loads are regular & coalesced.
// ---------------------------------------------------------------------------
__global__ void pack_w_kernel(const float* __restrict__ w_mass,
                              const float* __restrict__ w_decay,
                              float* __restrict__ wpack) {
    int idx = blockIdx.x * blockDim.x + threadIdx.x;   // 0 .. D*16-1
    int k = idx >> 4;
    int n = idx & 15;
    float v = 0.0f;
    if (n == 0) v = w_mass[k];
    else if (n == 1) v = w_decay[k];
    wpack[idx] = v;
}

// ---------------------------------------------------------------------------
// Kernel 1: gates via V_WMMA_F32_16X16X4_F32.
// One wave computes gates for 16 consecutive timesteps of one batch.
// D = A(16x4 slice of x) * B(4x16 packed weights) accumulated over K=D.
// A layout (32-bit 16x4): lanes 0-15 -> M=lane, K=0(v0),1(v1);
//                         lanes 16-31 -> M=lane-16, K=2(v0),3(v1).
// B layout (4x16):        lanes 0-15 -> N=lane, K=0(v0),1(v1);
//                         lanes 16-31 -> N=lane-16, K=2(v0),3(v1).
// C/D layout (16x16 f32): VGPR j: lanes 0-15 -> (M=j, N=lane),
//                                 lanes 16-31 -> (M=8+j, N=lane-16).
// => mass rows 0-7 in lane 0, rows 8-15 in lane 16; decay in lanes 1 / 17.
// gates stored interleaved: gates[2*(b*T+t)+0]=mass, +1=decay.
// ---------------------------------------------------------------------------
__global__ void gate_wmma_kernel(const float* __restrict__ x,
                                 const float* __restrict__ wpack,
                                 float* __restrict__ gates) {
    const int wave = (blockIdx.x * blockDim.x + threadIdx.x) >> 5;  // 0..2047
    const int lane = threadIdx.x & 31;
    const int tiles_per_b = TT / 16;            // 256
    const int b  = wave >> 8;                   // wave / 256
    const int t0 = (wave & (tiles_per_b - 1)) << 4;

    const int row  = lane & 15;                 // M (and N for B frag)
    const int koff = (lane >> 4) << 1;          // 0 or 2

    const float* aptr = x + ((size_t)(b * TT + t0 + row)) * DD + koff;
    const float* bptr = wpack + (size_t)koff * 16 + (lane & 15);

    v8f acc = {};
    #pragma unroll 4
    for (int kb = 0; kb < DD; kb += 4) {
        v2f a = *(const v2f*)(aptr + kb);       // x[row, kb+koff .. +1]
        v2f bv;
        bv.x = bptr[(size_t)kb * 16];           // B[kb+koff  ][n]
        bv.y = bptr[(size_t)kb * 16 + 16];      // B[kb+koff+1][n]
        acc = __builtin_amdgcn_wmma_f32_16x16x4_f32(
            /*neg_a=*/false, a, /*neg_b=*/false, bv,
            /*c_mod=*/(short)0, acc, /*reuse_a=*/false, /*reuse_b=*/false);
    }

    // Extract columns N=0 (mass) and N=1 (decay).
    if (lane == 0 || lane == 1 || lane == 16 || lane == 17) {
        const int rbase = (lane >= 16) ? 8 : 0;
        const int col   = lane & 15;            // 0 = mass, 1 = decay
        #pragma unroll
        for (int j = 0; j < 8; ++j) {
            float v = acc[j];
            v = 1.0f / (1.0f + __expf(-v));     // sigmoid
            gates[2u * ((size_t)b * TT + (size_t)(t0 + rbase + j)) + col] = v;
        }
    }
}

// ---------------------------------------------------------------------------
// Kernel 2: linear recurrence over T.  Each thread owns 2 consecutive d
// channels (float2): wave = 256B contiguous per step, fully coalesced.
// 8192 threads total = 256 waves.  Gate pair loads are block-uniform.
// global_prefetch issued 8 steps ahead (x steps stride 8KB in memory).
// ---------------------------------------------------------------------------
__global__ void scan_kernel(const float* __restrict__ x,
                            const float* __restrict__ gates,
                            float* __restrict__ out) {
    const int tid   = blockIdx.x * blockDim.x + threadIdx.x;  // 0..8191
    const int b     = tid >> 10;                // 1024 threads per batch
    const int dpair = tid & 1023;
    const size_t d0 = (size_t)dpair * 2;

    const float* xb = x   + (size_t)b * TT * DD + d0;
    float*       ob = out + (size_t)b * TT * DD + d0;
    const float* gb = gates + (size_t)b * TT * 2;

    float sx = 0.0f, sy = 0.0f;
    for (int t = 0; t < TT; ++t) {
        const float2 g  = *(const float2*)(gb + 2 * (size_t)t);      // (mass, decay)
        const float2 xv = *(const float2*)(xb + (size_t)t * DD);
        sx = g.y * sx + g.x * xv.x;
        sy = g.y * sy + g.x * xv.y;
        float2 s; s.x = sx; s.y = sy;
        *(float2*)(ob + (size_t)t * DD) = s;

        const int tp = (t + 8 < TT) ? (t + 8) : (TT - 1);
        __builtin_prefetch(xb + (size_t)tp * DD, 0, 0);              // global_prefetch_b8
    }
}

extern "C" void kernel_launch(void* const* d_in, const int* in_sizes, int n_in,
                              void* d_out, int out_size, void* d_ws, size_t ws_size,
                              hipStream_t stream) {
    const float* x       = (const float*)d_in[0];
    const float* w_mass  = (const float*)d_in[1];
    const float* w_decay = (const float*)d_in[2];
    float* out = (float*)d_out;

    // Workspace layout: [0, D*16) packed weights; then interleaved gates [B*T*2].
    float* wpack = (float*)d_ws;
    float* gates = wpack + (size_t)DD * 16;

    // 1) Pack weight B-matrix image: D*16 = 32768 elements.
    pack_w_kernel<<<(DD * 16) / 256, 256, 0, stream>>>(w_mass, w_decay, wpack);

    // 2) Gates: B*T/16 = 2048 tiles, 1 wave each, 8 waves / 256-thread block.
    gate_wmma_kernel<<<(BB * TT / 16) / 8, 256, 0, stream>>>(x, wpack, gates);

    // 3) Scan: 8192 threads, 64-thread blocks -> 128 blocks / 256 waves.
    scan_kernel<<<(BB * (DD / 2)) / 64, 64, 0, stream>>>(x, gates, out);
}